// SAN1d_1829656068699
// MI455X (gfx1250) — compile-verified
//
#include <hip/hip_runtime.h>
#include <hip/hip_bf16.h>

typedef float v2f __attribute__((ext_vector_type(2)));
typedef float v8f __attribute__((ext_vector_type(8)));

#define LLEN 16384
#define BATCH 256
#define NNEUR 8

static const size_t OFF_RECON = 0;
static const size_t OFF_SIM   = 4194304;          // 256*16384
static const size_t OFF_ACT   = 4194304 + 33554432;
static const size_t OFF_RECS  = 4194304 + 2u*33554432;

struct WPtrs { const float* w[8]; };

// ---------------------------------------------------------------------------
// Pack the 8 odd-sized kernels into a unified, zero-padded Wu[36][16] matrix:
// Wu[t][n] = w_n[t - 16 + pad_n] when in range, else 0.  (taps t in [0,32];
// rows 33..35 are zero padding so K=36 is a multiple of the WMMA K=4.)
// ---------------------------------------------------------------------------
__global__ void pack_weights(WPtrs wp, float* __restrict__ Wu) {
    const int KS[8] = {3, 5, 7, 9, 11, 15, 21, 33};
    for (int i = threadIdx.x; i < 36 * 16; i += blockDim.x) {
        int t = i >> 4;
        int n = i & 15;
        float v = 0.f;
        if (n < 8 && t < 33) {
            int Kn  = KS[n];
            int pad = (Kn - 1) >> 1;
            int j   = t - 16 + pad;
            if (j >= 0 && j < Kn) v = wp.w[n][j];
        }
        Wu[i] = v;
    }
}

// ---------------------------------------------------------------------------
// Kernel 1: similarity conv for ALL 8 neurons in one pass, via FP32 WMMA.
// C[M=16 positions][N=16 neurons] = A[M][K=36 taps] x B[K][N], 9 chunks of K=4.
// Block = 8 waves; each wave owns 16 consecutive positions; x window staged
// in LDS with zero halo.
// ---------------------------------------------------------------------------
__global__ void conv_sim_wmma(const float* __restrict__ x,
                              const float* __restrict__ Wu,
                              float* __restrict__ sim_base) {
    __shared__ float srow[176];          // [p0-16 .. p0+146] window (163 used)

    const int lane = threadIdx.x & 31;
    const int wv   = threadIdx.x >> 5;   // wave 0..7
    const int grp  = blockIdx.x;         // 256 samples * 128 groups
    const int b    = grp >> 7;
    const int p0   = (grp & 127) * 128;

    const float* xr = x + ((size_t)b << 14);
    for (int i = threadIdx.x; i < 176; i += 256) {
        int g = p0 - 16 + i;
        srow[i] = (g >= 0 && g < LLEN) ? xr[g] : 0.f;
    }
    __syncthreads();

    const int m     = lane & 15;              // M (positions) for A, N (neuron) for B
    const int khalf = (lane >> 4) << 1;       // 0 for lanes 0-15, 2 for 16-31

    // Loop-invariant B operands (Wu is [36][16] row-major).
    v2f bw[9];
#pragma unroll
    for (int c = 0; c < 9; ++c) {
        int k0  = c * 4 + khalf;              // <= 34
        bw[c][0] = Wu[k0 * 16 + m];
        bw[c][1] = Wu[(k0 + 1) * 16 + m];
    }

    v8f acc = {0.f, 0.f, 0.f, 0.f, 0.f, 0.f, 0.f, 0.f};
    const int baseidx = 16 * wv + m;          // LDS index of x[p0 + 16w + m - 16]
#pragma unroll
    for (int c = 0; c < 9; ++c) {
        int k0 = c * 4 + khalf;
        v2f a;
        a[0] = srow[baseidx + k0];
        a[1] = srow[baseidx + k0 + 1];
        acc = __builtin_amdgcn_wmma_f32_16x16x4_f32(
            false, a, false, bw[c], (short)0, acc, false, false);
    }

    // C layout: VGPR v, lanes 0-15 -> M=v, lanes 16-31 -> M=v+8; N = lane&15.
    if (m < NNEUR) {
        const int n    = m;
        const int moff = (lane >> 4) * 8;
        float* o = sim_base + (((size_t)(b * NNEUR + n)) << 14) + (p0 + 16 * wv + moff);
        float4 lo = make_float4(acc[0], acc[1], acc[2], acc[3]);
        float4 hi = make_float4(acc[4], acc[5], acc[6], acc[7]);
        *(float4*)(o)     = lo;
        *(float4*)(o + 4) = hi;
    }
}

// ---------------------------------------------------------------------------
// Kernel 2: per-(b,n) exact top-k(|v|) via 4-pass radix select on abs bits.
// Deterministic (integer LDS atomics only). Ties broken by smallest index
// using a stable chunked prefix scan (rare path).
// ---------------------------------------------------------------------------
__global__ void topk_kernel(const float* __restrict__ sim_base,
                            float* __restrict__ act_base) {
    __shared__ unsigned hist[256];
    __shared__ unsigned bitmap[512];      // 16384 bits
    __shared__ unsigned sbuf[256];
    __shared__ unsigned sh_base;

    const int row = blockIdx.x;           // b*8+n
    const int n   = row & 7;
    const unsigned k = 64u << (n & 3);    // 64,128,256,512 pattern
    const int tid = threadIdx.x;

    const float* sim_row = sim_base + ((size_t)row << 14);
    float*       act_row = act_base + ((size_t)row << 14);

    unsigned kk = k, prefix = 0, count_eq = 0;
    for (int s = 24; s >= 0; s -= 8) {
        hist[tid] = 0;
        __syncthreads();
        unsigned hm = (s == 24) ? 0u : (0xFFFFFFFFu << (s + 8));
        for (int i = tid; i < LLEN; i += 256) {
            unsigned key = __float_as_uint(sim_row[i]) & 0x7FFFFFFFu;
            if (((key ^ prefix) & hm) == 0u)
                atomicAdd(&hist[(key >> s) & 0xFF], 1u);
        }
        __syncthreads();
        // every thread computes the same selection (uniform)
        unsigned cum = 0; int sel = -1; unsigned selh = 0;
        for (int bin = 255; bin >= 0; --bin) {
            unsigned h = hist[bin];
            if (sel < 0 && cum + h >= kk) { sel = bin; selh = h; kk -= cum; }
            cum += h;
        }
        prefix |= ((unsigned)sel) << s;
        count_eq = selh;
        __syncthreads();                  // hist re-zeroed next pass
    }
    const unsigned T = prefix;            // k-th largest abs key
    const unsigned needed = kk;           // #elems == T to accept

    if (count_eq == needed) {             // common case: no bit-level ties to split
        for (int i = tid; i < LLEN; i += 256) {
            float v = sim_row[i];
            unsigned key = __float_as_uint(v) & 0x7FFFFFFFu;
            act_row[i] = (key >= T) ? v : 0.f;
        }
    } else {                              // stable earliest-index tie break
        for (int j = tid; j < 512; j += 256) bitmap[j] = 0;
        if (tid == 0) sh_base = 0;
        __syncthreads();
        for (int chunk = 0; chunk < 64; ++chunk) {
            int i = chunk * 256 + tid;
            unsigned key = __float_as_uint(sim_row[i]) & 0x7FFFFFFFu;
            unsigned flag = (key == T) ? 1u : 0u;
            sbuf[tid] = flag;
            __syncthreads();
            for (int off = 1; off < 256; off <<= 1) {   // inclusive scan
                unsigned t = (tid >= off) ? sbuf[tid - off] : 0u;
                __syncthreads();
                sbuf[tid] += t;
                __syncthreads();
            }
            unsigned incl  = sbuf[tid];
            unsigned total = sbuf[255];
            unsigned base  = sh_base;
            if (flag && (base + incl - 1u) < needed)
                atomicOr(&bitmap[i >> 5], 1u << (i & 31));
            __syncthreads();
            if (tid == 0) sh_base = base + total;
            __syncthreads();
        }
        for (int i = tid; i < LLEN; i += 256) {
            float v = sim_row[i];
            unsigned key = __float_as_uint(v) & 0x7FFFFFFFu;
            bool keep = (key > T) || ((bitmap[i >> 5] >> (i & 31)) & 1u);
            act_row[i] = keep ? v : 0.f;
        }
    }
}

// ---------------------------------------------------------------------------
// Kernel 3: reconstruction conv (dense 33-tap, deterministic) from LDS-staged
// activation row.  rec[p] = sum_t act[p + t - 16] * Wu[t][n].
// ---------------------------------------------------------------------------
__global__ void recon_conv(const float* __restrict__ act_base,
                           const float* __restrict__ Wu,
                           float* __restrict__ recs_base) {
    extern __shared__ float srow[];       // 16384 + 32 floats (zero halo)
    const int row = blockIdx.x;
    const int n   = row & 7;
    const int tid = threadIdx.x;

    const float* act = act_base + ((size_t)row << 14);
    for (int i = tid; i < LLEN + 32; i += 256) {
        int g = i - 16;
        srow[i] = (g >= 0 && g < LLEN) ? act[g] : 0.f;
    }
    __syncthreads();

    float wn[33];
#pragma unroll
    for (int t = 0; t < 33; ++t) wn[t] = Wu[t * 16 + n];

    float* out = recs_base + ((size_t)row << 14);
    for (int j = 0; j < 64; ++j) {
        int p = j * 256 + tid;
        float acc = 0.f;
#pragma unroll
        for (int t = 0; t < 33; ++t) acc = fmaf(srow[p + t], wn[t], acc);
        out[p] = acc;
    }
}

// ---------------------------------------------------------------------------
// Kernel 4: reconstructed[b][p] = sum_n reconstructions[b][n][p]
// ---------------------------------------------------------------------------
__global__ void sum_neurons(const float* __restrict__ recs_base,
                            float* __restrict__ out) {
    size_t i = (size_t)blockIdx.x * 256 + threadIdx.x;   // 4,194,304 total
    int b = (int)(i >> 14);
    int p = (int)(i & 16383);
    float s = 0.f;
#pragma unroll
    for (int n = 0; n < NNEUR; ++n)
        s += recs_base[(((size_t)(b * NNEUR + n)) << 14) + p];
    out[i] = s;
}

// ---------------------------------------------------------------------------
extern "C" void kernel_launch(void* const* d_in, const int* in_sizes, int n_in,
                              void* d_out, int out_size, void* d_ws, size_t ws_size,
                              hipStream_t stream) {
    const float* x  = (const float*)d_in[0];
    float* out      = (float*)d_out;
    float* Wu       = (float*)d_ws;       // 576 floats of scratch

    WPtrs wp;
    if (n_in >= 9) {
        for (int i = 0; i < 8; ++i) wp.w[i] = (const float*)d_in[1 + i];
    } else {                               // weights concatenated into one buffer
        static const int off[8] = {0, 3, 8, 15, 24, 35, 50, 71};
        const float* base = (const float*)d_in[1];
        for (int i = 0; i < 8; ++i) wp.w[i] = base + off[i];
    }

    pack_weights<<<1, 256, 0, stream>>>(wp, Wu);

    conv_sim_wmma<<<BATCH * (LLEN / 128), 256, 0, stream>>>(x, Wu, out + OFF_SIM);

    topk_kernel<<<BATCH * NNEUR, 256, 0, stream>>>(out + OFF_SIM, out + OFF_ACT);

    recon_conv<<<BATCH * NNEUR, 256, (LLEN + 32) * sizeof(float), stream>>>(
        out + OFF_ACT, Wu, out + OFF_RECS);

    sum_neurons<<<(BATCH * LLEN) / 256, 256, 0, stream>>>(out + OFF_RECS, out + OFF_RECON);
}